// EncoderClassifierMamba_53721450938847
// MI455X (gfx1250) — compile-verified
//
#include <hip/hip_runtime.h>
#include <hip/hip_bf16.h>

// B=64, S=37, T=2048, E=256
#define Bn 64
#define Sn 37
#define Tn 2048
#define En 256
#define F2S 74          // 2*S features
#define NQ 76           // 74 feature sums + count + time sum
#define TCHUNK 1024     // timesteps per block (256 threads x 4 t/thread)
#define NCHUNK 2        // Tn / TCHUNK

typedef float v2f __attribute__((ext_vector_type(2)));
typedef float v4f __attribute__((ext_vector_type(4)));
typedef float v8f __attribute__((ext_vector_type(8)));
typedef int   v4i __attribute__((ext_vector_type(4)));

__device__ __forceinline__ float waveReduceSum(float v) {
#pragma unroll
    for (int off = 16; off > 0; off >>= 1)
        v += __shfl_xor(v, off, 32);
    return v;
}

// ---------------------------------------------------------------------------
// Kernel 1: per (batch, t-chunk) partial reductions over T.
// Each thread owns 4 consecutive timesteps -> all streaming traffic is
// global_load_b128 (512 B per wave per instruction; maximizes bytes in
// flight per LOADcnt slot for the 23.3 TB/s HBM stream).
//   q in [0,37): sum_t m[t]*x[b,q,t]
//   q in [37,74): sum_t m[t]*sensor_mask[b,q-37,t]
//   q==74: sum_t m[t]   (count)     q==75: sum_t m[t]*time[b,t]
// ---------------------------------------------------------------------------
__global__ void k1_partial(const float* __restrict__ x,
                           const int* __restrict__ smask,
                           const float* __restrict__ timev,
                           float* __restrict__ part /* [B][NCHUNK][NQ] */) {
    const int b    = blockIdx.x;
    const int c    = blockIdx.y;
    const int tid  = threadIdx.x;          // 0..255
    const int t4   = c * TCHUNK + tid * 4; // first of this thread's 4 timesteps
    const int lane = tid & 31;
    const int wave = tid >> 5;             // 0..7

    const float* xb = x + (size_t)b * Sn * Tn;
    const int*   sb = smask + (size_t)b * Sn * Tn;

    // Pass A: per-timestep validity mask (any nonzero among 2S features)
    int f0 = 0, f1 = 0, f2 = 0, f3 = 0;
#pragma unroll
    for (int s = 0; s < Sn; ++s) {
        v4f xv = *(const v4f*)(xb + s * Tn + t4);
        v4i sv = *(const v4i*)(sb + s * Tn + t4);
        f0 |= (xv.x != 0.0f) | (sv.x != 0);
        f1 |= (xv.y != 0.0f) | (sv.y != 0);
        f2 |= (xv.z != 0.0f) | (sv.z != 0);
        f3 |= (xv.w != 0.0f) | (sv.w != 0);
    }
    const float m0 = f0 ? 1.0f : 0.0f;
    const float m1 = f1 ? 1.0f : 0.0f;
    const float m2 = f2 ? 1.0f : 0.0f;
    const float m3 = f3 ? 1.0f : 0.0f;

    __shared__ float wpart[NQ][8];

    // Pass B: masked sums per sensor. Re-reads hit WGP$/L2 (each block's
    // working set is ~300 KB of its own rows; 192 MB L2 holds everything).
#pragma unroll 1
    for (int s = 0; s < Sn; ++s) {
        v4f xv = *(const v4f*)(xb + s * Tn + t4);
        v4i sv = *(const v4i*)(sb + s * Tn + t4);
        float px = m0 * xv.x + m1 * xv.y + m2 * xv.z + m3 * xv.w;
        float pm = m0 * (float)sv.x + m1 * (float)sv.y
                 + m2 * (float)sv.z + m3 * (float)sv.w;
        px = waveReduceSum(px);
        pm = waveReduceSum(pm);
        if (lane == 0) { wpart[s][wave] = px; wpart[Sn + s][wave] = pm; }
    }
    v4f tv = *(const v4f*)(timev + (size_t)b * Tn + t4);
    float vc = waveReduceSum(m0 + m1 + m2 + m3);
    float vt = waveReduceSum(m0 * tv.x + m1 * tv.y + m2 * tv.z + m3 * tv.w);
    if (lane == 0) { wpart[74][wave] = vc; wpart[75][wave] = vt; }
    __syncthreads();

    if (tid < NQ) {
        float sum = 0.0f;
#pragma unroll
        for (int w = 0; w < 8; ++w) sum += wpart[tid][w];
        part[((size_t)b * NCHUNK + c) * NQ + tid] = sum;
    }
}

// ---------------------------------------------------------------------------
// Kernel 1b: fold chunk partials -> red[B][NQ]  (deterministic, no atomics)
// ---------------------------------------------------------------------------
__global__ void k1_reduce(const float* __restrict__ part,
                          float* __restrict__ red /* [B][NQ] */) {
    const int b = blockIdx.x;
    const int q = threadIdx.x;
    if (q < NQ) {
        float s = 0.0f;
#pragma unroll
        for (int c = 0; c < NCHUNK; ++c)
            s += part[((size_t)b * NCHUNK + c) * NQ + q];
        red[b * NQ + q] = s;
    }
}

// ---------------------------------------------------------------------------
// Kernel 2a: comb[B][2E]:
//   e<256 : pooled = (acc@W_sens + cnt*(b_sens+b_time) + tsum*W_time) / max(cnt,1e-9)
//   e>=256: static @ W_stat + b_stat
// Exact algebraic refactoring of the reference masked mean.
// ---------------------------------------------------------------------------
__global__ void k2_comb(const float* __restrict__ red,
                        const float* __restrict__ statics,
                        const float* __restrict__ W_sens, const float* __restrict__ b_sens,
                        const float* __restrict__ W_time, const float* __restrict__ b_time,
                        const float* __restrict__ W_stat, const float* __restrict__ b_stat,
                        float* __restrict__ comb /* [B][512] */) {
    const int idx = blockIdx.x * blockDim.x + threadIdx.x;  // 0..32767
    const int b = idx >> 9;
    const int e = idx & 511;
    const float* r = red + b * NQ;
    float val;
    if (e < En) {
        const float cnt = r[74];
        const float inv = 1.0f / fmaxf(cnt, 1e-9f);
        float sum = 0.0f;
#pragma unroll 2
        for (int f = 0; f < F2S; ++f) sum += r[f] * W_sens[f * En + e];
        sum += cnt * (b_sens[e] + b_time[e]);
        sum += r[75] * W_time[e];
        val = sum * inv;
    } else {
        const int e2 = e - En;
        float sum = b_stat[e2];
#pragma unroll
        for (int j = 0; j < 8; ++j) sum += statics[b * 8 + j] * W_stat[j * En + e2];
        val = sum;
    }
    comb[b * 512 + e] = val;
}

// ---------------------------------------------------------------------------
// Kernel 2b: merged = relu(comb[64,512] @ W_merge[512,512] + b_merge)
// via V_WMMA_F32_16X16X4_F32 (exact f32 math). One wave per 16x16 tile.
// A frag (16x4): lane = m%16 + 16*(k>>1), vgpr = k&1
// B frag (4x16): lane = n%16 + 16*(k>>1), vgpr = k&1
// C/D (16x16):   vgpr v -> row v + 8*(lane>=16), col = lane&15
// ---------------------------------------------------------------------------
__global__ void k2_merge(const float* __restrict__ comb,
                         const float* __restrict__ W_merge,
                         const float* __restrict__ b_merge,
                         float* __restrict__ merged /* [64][512] */) {
    const int lane  = threadIdx.x;       // 0..31 (wave32)
    const int ntile = blockIdx.x;        // 0..31
    const int mtile = blockIdx.y;        // 0..3
    const int half  = lane >> 4;         // which K half this lane supplies
    const int l16   = lane & 15;
    const int m = mtile * 16 + l16;      // A row for this lane
    const int n = ntile * 16 + l16;      // B col for this lane

    v8f c = {};
    for (int k = 0; k < 512; k += 4) {
        const int kk = k + 2 * half;
        v2f a = *(const v2f*)(comb + (size_t)m * 512 + kk);   // contiguous in K
        v2f bm;
        bm.x = W_merge[(size_t)kk * 512 + n];
        bm.y = W_merge[(size_t)(kk + 1) * 512 + n];
        c = __builtin_amdgcn_wmma_f32_16x16x4_f32(
                /*neg_a=*/false, a, /*neg_b=*/false, bm,
                /*c_mod=*/(short)0, c, /*reuse_a=*/false, /*reuse_b=*/false);
    }

    const float bias = b_merge[n];       // bias indexed by this lane's column
#pragma unroll
    for (int v = 0; v < 8; ++v) {
        const int row = mtile * 16 + v + 8 * half;
        const int col = ntile * 16 + l16;
        merged[(size_t)row * 512 + col] = fmaxf(c[v] + bias, 0.0f);
    }
}

// ---------------------------------------------------------------------------
// Kernel 2c: out[64,2] = merged @ W_cls + b_cls
// ---------------------------------------------------------------------------
__global__ void k2_cls(const float* __restrict__ merged,
                       const float* __restrict__ W_cls,
                       const float* __restrict__ b_cls,
                       float* __restrict__ out) {
    const int idx = blockIdx.x * blockDim.x + threadIdx.x;  // 0..127
    if (idx >= Bn * 2) return;
    const int b = idx >> 1;
    const int cls = idx & 1;
    float s = b_cls[cls];
#pragma unroll 4
    for (int k = 0; k < 512; ++k) s += merged[b * 512 + k] * W_cls[k * 2 + cls];
    out[b * 2 + cls] = s;
}

extern "C" void kernel_launch(void* const* d_in, const int* in_sizes, int n_in,
                              void* d_out, int out_size, void* d_ws, size_t ws_size,
                              hipStream_t stream) {
    const float* x       = (const float*)d_in[0];   // [64,37,2048]
    const float* statics = (const float*)d_in[1];   // [64,8]
    const float* timev   = (const float*)d_in[2];   // [64,2048]
    const int*   smask   = (const int*)  d_in[3];   // [64,37,2048]
    const float* W_sens  = (const float*)d_in[4];   // [74,256]
    const float* b_sens  = (const float*)d_in[5];   // [256]
    const float* W_time  = (const float*)d_in[6];   // [1,256]
    const float* b_time  = (const float*)d_in[7];   // [256]
    const float* W_stat  = (const float*)d_in[8];   // [8,256]
    const float* b_stat  = (const float*)d_in[9];   // [256]
    const float* W_merge = (const float*)d_in[10];  // [512,512]
    const float* b_merge = (const float*)d_in[11];  // [512]
    const float* W_cls   = (const float*)d_in[12];  // [512,2]
    const float* b_cls   = (const float*)d_in[13];  // [2]
    float* out = (float*)d_out;                     // [64,2]

    // Workspace carve-up (floats)
    float* ws     = (float*)d_ws;
    float* part   = ws;                                  // 64*2*76 = 9728
    float* red    = part + (size_t)Bn * NCHUNK * NQ;     // 64*76   = 4864
    float* comb   = red + (size_t)Bn * NQ;               // 64*512  = 32768
    float* merged = comb + (size_t)Bn * 512;             // 64*512  = 32768

    k1_partial<<<dim3(Bn, NCHUNK), 256, 0, stream>>>(x, smask, timev, part);
    k1_reduce<<<Bn, 128, 0, stream>>>(part, red);
    k2_comb<<<(Bn * 512) / 256, 256, 0, stream>>>(red, statics, W_sens, b_sens,
                                                  W_time, b_time, W_stat, b_stat, comb);
    k2_merge<<<dim3(32, 4), 32, 0, stream>>>(comb, W_merge, b_merge, merged);
    k2_cls<<<1, 128, 0, stream>>>(merged, W_cls, b_cls, out);
}